// MXQuantizer_58557584113816
// MI455X (gfx1250) — compile-verified
//
#include <hip/hip_runtime.h>
#include <stdint.h>

#define WAVE          32
#define BLK           128      // quantization block (floats)
#define BPW           8        // blocks streamed per wave
#define WAVES_PER_WG  8        // 256-thread workgroup

typedef float v4f __attribute__((ext_vector_type(4)));   // native vector for b128 ops

// ---------- exact floor(log2(x)) for x > 0 via exponent bits ----------
__device__ __forceinline__ int ifloor_log2(float ax) {
  unsigned b = __float_as_uint(ax);
  int e = (int)((b >> 23) & 0xFFu);
  if (e != 0) return e - 127;                 // normal
  b = __float_as_uint(ax * 16777216.0f);      // denormal: * 2^24, re-extract
  return (int)((b >> 23) & 0xFFu) - 151;      // -127 - 24
}

// reference rounding: sign(t) * floor(|t| + 0.5)
__device__ __forceinline__ float round_half_away(float t) {
  float r = floorf(fabsf(t) + 0.5f);
  return t < 0.0f ? -r : r;
}

// wave32 butterfly via ds_swizzle group-of-32 xor mode:
// offset[14:10]=xor_mask, offset[9:5]=or_mask(0), offset[4:0]=and_mask(0x1f)
template <int XORMASK>
__device__ __forceinline__ float swz_xor(float v) {
  int r = __builtin_amdgcn_ds_swizzle(__float_as_int(v), (XORMASK << 10) | 0x1f);
  return __int_as_float(r);
}

__device__ __forceinline__ float wave_sum(float v) {
  v += swz_xor<16>(v);
  v += swz_xor<8>(v);
  v += swz_xor<4>(v);
  v += swz_xor<2>(v);
  v += swz_xor<1>(v);
  return v;
}
__device__ __forceinline__ float wave_max(float v) {
  v = fmaxf(v, swz_xor<16>(v));
  v = fmaxf(v, swz_xor<8>(v));
  v = fmaxf(v, swz_xor<4>(v));
  v = fmaxf(v, swz_xor<2>(v));
  v = fmaxf(v, swz_xor<1>(v));
  return v;
}

// int4-ish inlier: ebits=0, mbits=6 -> scale 2^4, max_norm 31/16
__device__ __forceinline__ float quant_inlier(float v /* already / s_in */) {
  float q = round_half_away(v * 16.0f) * 0.0625f;
  return fminf(fmaxf(q, -1.9375f), 1.9375f);
}

// fp8 e4m3 outlier: ebits=4, mbits=5, max_norm 448, pexp floor at -6
__device__ __forceinline__ float quant_outlier(float A /* already / s_out */) {
  int p = (A == 0.0f) ? 0 : ifloor_log2(fabsf(A));
  p = (p < -6) ? -6 : p;
  float q = round_half_away(A * ldexpf(1.0f, 3 - p)) * ldexpf(1.0f, p - 3);
  return fminf(fmaxf(q, -448.0f), 448.0f);
}

__global__ __launch_bounds__(WAVE * WAVES_PER_WG)
void mxq_kernel(const float* __restrict__ x, float* __restrict__ out, int nblocks) {
  const int lane = threadIdx.x & (WAVE - 1);
  const int wv   = threadIdx.x >> 5;
  const int wave_global = blockIdx.x * WAVES_PER_WG + wv;
  const int base = wave_global * BPW;

  int cnt = nblocks - base;           // wave-uniform
  if (cnt <= 0) return;
  if (cnt > BPW) cnt = BPW;

  __shared__ v4f smem[2][WAVES_PER_WG][WAVE];
  unsigned lds_addr[2];
  lds_addr[0] = (unsigned)(uintptr_t)(&smem[0][wv][lane]);
  lds_addr[1] = (unsigned)(uintptr_t)(&smem[1][wv][lane]);

  // CDNA5 async DMA: copies global saddr + vaddr into LDS at vdst; tracked by ASYNCcnt.
  // NT hint: 128MB streamed exactly once.
  auto issue = [&](int bid, unsigned lds) {
    int voff = bid * (BLK * 4) + lane * 16;   // byte offset, GVS mode
    asm volatile("global_load_async_to_lds_b128 %0, %1, %2 th:TH_LOAD_NT"
                 :
                 : "v"(lds), "v"(voff), "s"(x)
                 : "memory");
  };

  issue(base, lds_addr[0]);                   // prologue fill

  const float MINN = 1.17549435e-38f;         // FP32_MIN_NORMAL

  for (int i = 0; i < cnt; ++i) {
    const bool more = (i + 1 < cnt);
    if (more) issue(base + i + 1, lds_addr[(i + 1) & 1]);

    // async loads complete in order: <=1 outstanding means block i is in LDS
    if (more) asm volatile("s_wait_asynccnt 0x1" ::: "memory");
    else      asm volatile("s_wait_asynccnt 0x0" ::: "memory");

    v4f a = smem[i & 1][wv][lane];            // ds_load_b128
    float av[4] = {a.x, a.y, a.z, a.w};
    float ax[4] = {fabsf(a.x), fabsf(a.y), fabsf(a.z), fabsf(a.w)};

    // block mean / population std of |A| (two-pass, like the reference)
    float mean = wave_sum(ax[0] + ax[1] + ax[2] + ax[3]) * 0.0078125f;
    float d0 = ax[0] - mean, d1 = ax[1] - mean, d2 = ax[2] - mean, d3 = ax[3] - mean;
    float var = wave_sum(d0 * d0 + d1 * d1 + d2 * d2 + d3 * d3) * 0.0078125f;
    float sd  = sqrtf(var);
    float lo = mean - 2.0f * sd, hi = mean + 2.0f * sd;

    float inl[4], oul[4];
    float mi = 0.0f, mo = 0.0f;
#pragma unroll
    for (int k = 0; k < 4; ++k) {
      bool o = (av[k] < lo) || (av[k] > hi);
      inl[k] = o ? 0.0f : av[k];
      oul[k] = o ? av[k] : 0.0f;
      mi = fmaxf(mi, fabsf(inl[k]));
      mo = fmaxf(mo, fabsf(oul[k]));
    }
    mi = wave_max(mi);
    mo = wave_max(mo);

    // inlier shared exponent (EMAX_IN = 0), clamp se < -127 -> -20
    int   se_i  = ifloor_log2(mi == 0.0f ? MINN : mi);
    float sef_i = (se_i < -127) ? -20.0f : (float)se_i;
    int   ei    = (int)sef_i;
    float s_in     = ldexpf(1.0f, ei);
    float s_in_inv = ldexpf(1.0f, -ei);

    // outlier shared exponent on (outlier * s_in), EMAX_OUT = 8
    float mov   = mo * s_in;
    int   se_o  = ifloor_log2(mov == 0.0f ? MINN : mov) - 8;
    float sef_o = (se_o < -127) ? -20.0f : (float)se_o;
    int   eo    = (int)sef_o;
    float s_o     = ldexpf(1.0f, eo);
    float s_o_inv = ldexpf(1.0f, -eo);

    v4f r;
#pragma unroll
    for (int k = 0; k < 4; ++k) {
      float iq = quant_inlier(inl[k] * s_in_inv) * s_in;
      float oq = quant_outlier((oul[k] * s_in) * s_o_inv) * s_o * s_in_inv;
      r[k] = iq + oq;
    }

    v4f* op = (v4f*)(out + (size_t)(base + i) * BLK + (size_t)lane * 4);
    __builtin_nontemporal_store(r, op);        // global_store_b128 th:TH_STORE_NT
  }
}

extern "C" void kernel_launch(void* const* d_in, const int* in_sizes, int n_in,
                              void* d_out, int out_size, void* d_ws, size_t ws_size,
                              hipStream_t stream) {
  (void)n_in; (void)d_ws; (void)ws_size; (void)out_size;
  const float* x = (const float*)d_in[0];
  float* out = (float*)d_out;

  long long n = in_sizes[0];
  int nblocks = (int)(n / BLK);
  if (nblocks <= 0) return;

  int waves = (nblocks + BPW - 1) / BPW;
  int wgs   = (waves + WAVES_PER_WG - 1) / WAVES_PER_WG;

  mxq_kernel<<<dim3(wgs), dim3(WAVE * WAVES_PER_WG), 0, stream>>>(x, out, nblocks);
}